// SkipgramNegSampling_46952582480430
// MI455X (gfx1250) — compile-verified
//
#include <hip/hip_runtime.h>

typedef __attribute__((ext_vector_type(2))) float v2f;
typedef __attribute__((ext_vector_type(8))) float v8f;

#define VOCAB   100000
#define DIM     128
#define BATCH   16384
#define KNEG    20
#define TILES   (BATCH / 16)            // 1024 tiles of 16 batch rows
#define WPB     8                       // waves per block
#define BLOCKS  (TILES / WPB)           // 128 blocks of 256 threads

__device__ __forceinline__ float log_sigmoid(float x) {
    // stable: min(x,0) - log1p(exp(-|x|))
    return fminf(x, 0.0f) - log1pf(expf(-fabsf(x)));
}

__global__ void __launch_bounds__(256)
sgns_score_kernel(const int*   __restrict__ center,
                  const int*   __restrict__ target,
                  const int*   __restrict__ negs,
                  const float* __restrict__ V,
                  const float* __restrict__ U,
                  float*       __restrict__ partials)
{
    const int lane = threadIdx.x & 31;
    const int wave = threadIdx.x >> 5;
    const int tile = blockIdx.x * WPB + wave;
    const int b0   = tile * 16;

    // WMMA f32 16x16x4 A-operand layout: lane L holds row M=L%16,
    // K-pair at offset 2*(L/16). Same layout serves the B-operand of
    // the transposed matrix, so CE and U rows load identically.
    const int j = lane & 15;            // matrix row this lane feeds
    const int h = (lane >> 4) * 2;      // k offset within each 4-chunk

    // ---- load CE tile (16x128) into A-operand registers, 32 K-chunks ----
    const int    crow = center[b0 + j];
    const float* cptr = V + (size_t)crow * DIM + h;
    v2f ceA[32];
#pragma unroll
    for (int kc = 0; kc < 32; ++kc)
        ceA[kc] = *(const v2f*)(cptr + 4 * kc);

    // diagonal ownership: m=0..7 -> lane m (VGPR m); m=8..15 -> lane m+16 (VGPR m-8)
    const bool active = (lane < 8) || (lane >= 24);
    const int  didx   = (lane < 8) ? lane : (lane - 24);

    float partial = 0.0f;

    for (int s = 0; s <= KNEG; ++s) {   // s=0: positive (target), s>=1: negatives
        const int row = (s == 0) ? target[b0 + j]
                                 : negs[(b0 + j) * KNEG + (s - 1)];
        const float* uptr = U + (size_t)row * DIM + h;

        v8f acc = {};                   // S = CE (16x128) x Rows^T (128x16)
#pragma unroll
        for (int kc = 0; kc < 32; ++kc) {
            v2f bB = *(const v2f*)(uptr + 4 * kc);
            acc = __builtin_amdgcn_wmma_f32_16x16x4_f32(
                /*neg_a=*/false, ceA[kc],
                /*neg_b=*/false, bB,
                /*c_mod=*/(short)0, acc,
                /*reuse_a=*/false, /*reuse_b=*/false);
        }

        // branchless diagonal extraction (EXEC must stay all-1s around WMMA)
        float d = acc[0];
        d = (didx == 1) ? acc[1] : d;
        d = (didx == 2) ? acc[2] : d;
        d = (didx == 3) ? acc[3] : d;
        d = (didx == 4) ? acc[4] : d;
        d = (didx == 5) ? acc[5] : d;
        d = (didx == 6) ? acc[6] : d;
        d = (didx == 7) ? acc[7] : d;

        const float x = (s == 0) ? d : -d;
        const float c = log_sigmoid(x);
        partial += active ? c : 0.0f;
    }

    // wave32 butterfly reduction
#pragma unroll
    for (int off = 16; off > 0; off >>= 1)
        partial += __shfl_xor(partial, off, 32);

    if (lane == 0) partials[tile] = partial;
}

__global__ void __launch_bounds__(256)
sgns_reduce_kernel(const float* __restrict__ partials, float* __restrict__ out)
{
    __shared__ float sm[256];
    float s = 0.0f;
    for (int i = threadIdx.x; i < TILES; i += 256) s += partials[i];
    sm[threadIdx.x] = s;
    __syncthreads();
    for (int stride = 128; stride > 0; stride >>= 1) {
        if ((int)threadIdx.x < stride) sm[threadIdx.x] += sm[threadIdx.x + stride];
        __syncthreads();
    }
    if (threadIdx.x == 0) out[0] = -sm[0] / (float)BATCH;
}

extern "C" void kernel_launch(void* const* d_in, const int* in_sizes, int n_in,
                              void* d_out, int out_size, void* d_ws, size_t ws_size,
                              hipStream_t stream)
{
    const int*   center = (const int*)d_in[0];
    const int*   target = (const int*)d_in[1];
    const int*   negs   = (const int*)d_in[2];
    const float* V      = (const float*)d_in[3];
    const float* U      = (const float*)d_in[4];

    float* partials = (float*)d_ws;     // 1024 floats of scratch

    sgns_score_kernel<<<BLOCKS, 256, 0, stream>>>(center, target, negs, V, U, partials);
    sgns_reduce_kernel<<<1, 256, 0, stream>>>(partials, (float*)d_out);
}